// RGCN_26792005992743
// MI455X (gfx1250) — compile-verified
//
#include <hip/hip_runtime.h>

#define D 128
#define RNUM 4
#define LNUM 3
#define NEG_SLOPE 0.01f

typedef __attribute__((ext_vector_type(2))) float v2f;
typedef __attribute__((ext_vector_type(8))) float v8f;

// ---------------------------------------------------------------------------
// x[i,:] = emb[node_type[i],:]
// ---------------------------------------------------------------------------
__global__ void rgcn_embed(const int* __restrict__ node_type,
                           const float* __restrict__ emb,
                           float* __restrict__ x, int n) {
  int idx = blockIdx.x * blockDim.x + threadIdx.x;   // over n * D/4
  if (idx >= n * (D / 4)) return;
  int node = idx / (D / 4);
  int c4   = idx % (D / 4);
  const float4* e = (const float4*)(emb + (long)node_type[node] * D);
  ((float4*)(x + (long)node * D))[c4] = e[c4];
}

// ---------------------------------------------------------------------------
// Fused per-layer transform: for one layer l,
//   H[r] = x @ W[l,r]   (r = 0..3)
//   x    = x @ root[l] + bias[l]     (in-place, done last)
// One wave owns a 16-row A tile, kept in registers across all 5 matmuls.
// Each 128x128 B matrix is staged once per block into 64KB LDS as k-pair
// interleaved float2 so the WMMA B operand is a single conflict-free
// ds_load_b64. fp32 WMMA (V_WMMA_F32_16X16X4_F32), K=128 chained as 32 steps.
// ---------------------------------------------------------------------------
__global__ __launch_bounds__(256) void rgcn_gemm5(
    const float* __restrict__ A,       // x  [nRows, D]
    const float* __restrict__ Wl,      // W[l] : 4 contiguous DxD matrices
    const float* __restrict__ rootl,   // root[l] : DxD
    const float* __restrict__ bias,    // bias[l] : D
    float* __restrict__ H,             // [4, nRows, D]
    float* __restrict__ x,             // output for root transform (== A)
    int nRows) {
  __shared__ v2f Bs[64 * 128];         // Bs[p*128+n] = (B[2p][n], B[2p+1][n])

  const int wave = threadIdx.x >> 5;
  const int lane = threadIdx.x & 31;
  const int tile = blockIdx.x * 8 + wave;
  const int numTiles = nRows >> 4;     // N assumed multiple of 16
  const bool active = tile < numTiles; // wave-uniform: EXEC stays all-ones

  const int m  = lane & 15;            // A row within tile / output column lane
  const int kh = lane >> 4;            // which half-wave (k pair selector)
  const int kk = kh << 1;              // 0 or 2
  const long rowBase = (long)tile * 16;

  // Preload the wave's whole 16x128 A tile into registers (reused 5x).
  v2f a[32];
  if (active) {
    const float* arow = A + (rowBase + m) * D;
#pragma unroll
    for (int ks = 0; ks < 32; ++ks)
      a[ks] = *(const v2f*)(arow + 4 * ks + kk);   // 8B-aligned global_load_b64
  }

  for (int mat = 0; mat < 5; ++mat) {
    const float* B = (mat < 4) ? (Wl + (size_t)mat * D * D) : rootl;

    __syncthreads();                   // previous matrix's LDS reads complete
    // Cooperative fill: 256 threads x 32 float2 each (coalesced global reads,
    // conflict-free ds_store_b64).
    for (int j = threadIdx.x; j < 64 * 128; j += 256) {
      const int p = j >> 7, n = j & 127;
      v2f pr;
      pr.x = B[(size_t)(2 * p)     * D + n];
      pr.y = B[(size_t)(2 * p + 1) * D + n];
      Bs[j] = pr;
    }
    __syncthreads();
    if (!active) continue;             // wave-uniform; barrier counts match

    float* Cbase = (mat < 4) ? (H + (size_t)mat * nRows * D) : x;
    for (int nt = 0; nt < 8; ++nt) {
      const int ncol = nt * 16 + m;
      const float binit = (mat == 4) ? bias[ncol] : 0.0f;
      v8f acc = {binit, binit, binit, binit, binit, binit, binit, binit};
#pragma unroll
      for (int ks = 0; ks < 32; ++ks) {
        const v2f b = Bs[(2 * ks + kh) * 128 + ncol];   // ds_load_b64
        acc = __builtin_amdgcn_wmma_f32_16x16x4_f32(
            false, a[ks], false, b, (short)0, acc, false, false);
      }
      // C/D layout: VGPR v -> row = v + 8*kh, col = ncol
      float* crow = Cbase + (rowBase + kh * 8) * D + ncol;
#pragma unroll
      for (int v = 0; v < 8; ++v) crow[v * D] = acc[v];
    }
  }
}

// ---------------------------------------------------------------------------
// For each edge e: agg[dst[e], :] += H[edge_type[e]][src[e], :]
// 32 lanes per edge, 4 floats per lane, f32 atomics (L2-resident accumulator).
// ---------------------------------------------------------------------------
__global__ void rgcn_scatter(const int* __restrict__ edge_index,
                             const int* __restrict__ edge_type,
                             const float* __restrict__ H,
                             float* __restrict__ agg,
                             int nNodes, int nEdges) {
  long idx = (long)blockIdx.x * blockDim.x + threadIdx.x;
  int e = (int)(idx >> 5);
  if (e >= nEdges) return;
  int c   = ((int)idx & 31) * 4;
  int r   = edge_type[e];
  int src = edge_index[e];
  int dst = edge_index[nEdges + e];
  const float4 v = *(const float4*)(H + ((long)r * nNodes + src) * D + c);
  float* out = agg + (long)dst * D + c;
  atomicAdd(out + 0, v.x);
  atomicAdd(out + 1, v.y);
  atomicAdd(out + 2, v.z);
  atomicAdd(out + 3, v.w);
}

// ---------------------------------------------------------------------------
// out = leaky_relu(in), float4 vectorized
// ---------------------------------------------------------------------------
__global__ void rgcn_leaky(const float4* __restrict__ in,
                           float4* __restrict__ out, long n4) {
  long i = (long)blockIdx.x * blockDim.x + threadIdx.x;
  if (i >= n4) return;
  float4 v = in[i];
  v.x = v.x > 0.0f ? v.x : NEG_SLOPE * v.x;
  v.y = v.y > 0.0f ? v.y : NEG_SLOPE * v.y;
  v.z = v.z > 0.0f ? v.z : NEG_SLOPE * v.z;
  v.w = v.w > 0.0f ? v.w : NEG_SLOPE * v.w;
  out[i] = v;
}

extern "C" void kernel_launch(void* const* d_in, const int* in_sizes, int n_in,
                              void* d_out, int out_size, void* d_ws, size_t ws_size,
                              hipStream_t stream) {
  const int*   node_type  = (const int*)d_in[0];
  const int*   edge_index = (const int*)d_in[1];
  const int*   edge_type  = (const int*)d_in[2];
  const float* emb        = (const float*)d_in[3];
  const float* W          = (const float*)d_in[4];   // [L,R,D,D]
  const float* root       = (const float*)d_in[5];   // [L,D,D]
  const float* bias       = (const float*)d_in[6];   // [L,D]

  const int N = in_sizes[0];
  const int E = in_sizes[2];

  // workspace: H[R][N][D] then x[N][D]  (5*N*D floats = 128 MB at this size)
  float* H = (float*)d_ws;
  float* x = H + (size_t)RNUM * N * D;
  float* out = (float*)d_out;

  const int embThreads = N * (D / 4);
  rgcn_embed<<<(embThreads + 255) / 256, 256, 0, stream>>>(node_type, emb, x, N);

  const int tiles = N >> 4;
  dim3 gemmGrid((tiles + 7) / 8), gemmBlk(256);      // 8 waves/block, 1 tile/wave
  const long scatThreads = (long)E * 32;
  const int  scatBlocks  = (int)((scatThreads + 255) / 256);
  const long n4 = (long)N * D / 4;
  const int  eltBlocks = (int)((n4 + 255) / 256);

  for (int l = 0; l < LNUM; ++l) {
    // H_r = x @ W[l,r] for r=0..3, then x = x @ root[l] + bias[l] (fused)
    rgcn_gemm5<<<gemmGrid, gemmBlk, 0, stream>>>(
        x, W + (size_t)l * RNUM * D * D, root + (size_t)l * D * D,
        bias + (size_t)l * D, H, x, N);
    // x[dst] += H[rel][src] over all edges
    rgcn_scatter<<<scatBlocks, 256, 0, stream>>>(edge_index, edge_type, H, x, N, E);
    // activation; last layer writes final output
    float4* reluOut = (l == LNUM - 1) ? (float4*)out : (float4*)x;
    rgcn_leaky<<<eltBlocks, 256, 0, stream>>>((const float4*)x, reluOut, n4);
  }
}